// CTCLoss_1924145349160
// MI455X (gfx1250) — compile-verified
//
#include <hip/hip_runtime.h>
#include <stdint.h>

// ---------------- problem constants (from reference) ----------------
#define Bb    32
#define Tt    1024
#define Vv    2048
#define LMAXc 128
#define S_EXT (2 * LMAXc + 1)   // 257 extended symbols
#define BLANK (Vv - 1)
#define NEGF  (-1e30f)

// ---------------- logZ/lp kernel tiling ----------------
#define ROWS 4                   // rows per TDM tile (4 * 2048 * 4B = 32 KB)
#define CHUNKS_PER_BLOCK 8       // 32 rows per block -> all rows share one batch b
#define LOGZ_BLOCKS ((Bb * Tt) / (ROWS * CHUNKS_PER_BLOCK))  // 1024 blocks

typedef __attribute__((ext_vector_type(4))) unsigned int v4u;
typedef __attribute__((ext_vector_type(8))) int         v8i;
typedef __attribute__((ext_vector_type(4))) int         v4i;

// Low 32 bits of a flat LDS-aperture address are the LDS byte offset
// (ISA 10.2).
__device__ __forceinline__ unsigned lds_off(const void* p) {
  return (unsigned)(uintptr_t)p;
}

// Issue one TDM 2-D tile load: `nrows` consecutive rows of 2048 f32 from the
// activation tensor (viewed as [B*T, V]) into LDS at `ldsByte`.
// D# layout per CDNA5 ISA §8.3-8.6 (groups 2/3 zero -> 2-D tensor).
// clang-23 toolchain: 6-arg builtin (extra int32x8 operand, zero-filled).
__device__ __forceinline__ void tdm_load_rows(const float* gsrc,
                                              unsigned ldsByte, int nrows) {
  uint64_t ga = (uint64_t)(uintptr_t)gsrc;
  v4u g0;
  g0[0] = 1u;                                        // count=1, user descriptor
  g0[1] = ldsByte;                                   // lds_addr
  g0[2] = (unsigned)(ga & 0xFFFFFFFFu);              // global_addr[31:0]
  g0[3] = (unsigned)((ga >> 32) & 0x01FFFFFFu)       // global_addr[56:32]
          | (2u << 30);                              // type=2 (image)
  v8i g1;
  g1[0] = (int)(2u << 16);                           // data_size=4B; mask=0
  g1[1] = (int)((unsigned)(Vv & 0xFFFF) << 16);      // tensor_dim0[15:0]
  g1[2] = (int)(((unsigned)((Bb * Tt) & 0xFFFF) << 16)   // tensor_dim1[15:0]
                | (((unsigned)Vv >> 16) & 0xFFFF));      // tensor_dim0[31:16]
  g1[3] = (int)(((unsigned)(Vv & 0xFFFF) << 16)          // tile_dim0
                | (((unsigned)(Bb * Tt) >> 16) & 0xFFFF));
  g1[4] = (int)((unsigned)nrows & 0xFFFF);           // tile_dim1 (tile_dim2=0)
  g1[5] = (int)Vv;                                   // tensor_dim0_stride[31:0]
  g1[6] = 0;
  g1[7] = 0;
  v4i g2 = {0, 0, 0, 0};
  v4i g3 = {0, 0, 0, 0};
  v8i g4 = {0, 0, 0, 0, 0, 0, 0, 0};
  __builtin_amdgcn_tensor_load_to_lds(g0, g1, g2, g3, g4, /*cpol=*/0);
}

__device__ __forceinline__ float lae2(float a, float b) {
  float m = fmaxf(a, b);
  float n = fminf(a, b);
  return m + log1pf(__expf(n - m));   // safe for a==b==NEGF (adds ln2)
}

// ---------------------------------------------------------------------------
// Kernel A: for each (b,t) row: logZ = logsumexp over V (wave32 shfl tree),
// then, while the row sits in LDS, scatter-gather lp[r*S+s] = x[ext[s]] - logZ.
// Rows are staged HBM->LDS by the Tensor Data Mover, double-buffered.
// block = 128 threads (4 waves), one wave per row of the 4-row tile.
// ---------------------------------------------------------------------------
__global__ void __launch_bounds__(128)
ctc_logz_lp_kernel(const float* __restrict__ act,
                   const int* __restrict__ labels,
                   float* __restrict__ lp) {
  __shared__ float tile[2][ROWS][Vv];   // 64 KB double buffer
  __shared__ int ext[S_EXT];

  const int tid  = threadIdx.x;
  const int wave = tid >> 5;
  const int lane = tid & 31;
  const int row0 = blockIdx.x * (ROWS * CHUNKS_PER_BLOCK);  // 32 rows/block
  const int b    = row0 / Tt;           // T=1024 % 32 == 0 -> single b per block

  for (int s = tid; s < S_EXT; s += 128)
    ext[s] = (s & 1) ? labels[b * LMAXc + (s >> 1)] : BLANK;

  // preload chunk 0 (wave 0 issues; TDM ignores EXEC, one issue per wave)
  if (tid < 32)
    tdm_load_rows(act + (size_t)row0 * Vv, lds_off(&tile[0][0][0]), ROWS);

  for (int c = 0; c < CHUNKS_PER_BLOCK; ++c) {
    const int buf = c & 1;
    if (tid < 32) {
      if (c + 1 < CHUNKS_PER_BLOCK) {
        tdm_load_rows(act + (size_t)(row0 + (c + 1) * ROWS) * Vv,
                      lds_off(&tile[buf ^ 1][0][0]), ROWS);
        __builtin_amdgcn_s_wait_tensorcnt(1);  // oldest (current chunk) done
      } else {
        __builtin_amdgcn_s_wait_tensorcnt(0);
      }
    }
    __syncthreads();   // tile[buf] valid for all waves; ext valid (c==0)

    const float* row = &tile[buf][wave][0];
    // pass 1: row max (wave32 shfl-xor tree -> all lanes hold result)
    float m = NEGF;
    for (int k = lane; k < Vv; k += 32) m = fmaxf(m, row[k]);
    for (int off = 16; off; off >>= 1) m = fmaxf(m, __shfl_xor(m, off, 32));
    // pass 2: sum exp(x - m)
    float s = 0.0f;
    for (int k = lane; k < Vv; k += 32) s += __expf(row[k] - m);
    for (int off = 16; off; off >>= 1) s += __shfl_xor(s, off, 32);
    const float lz = m + __logf(s);

    // fused gather: lp for the 257 extended symbols of this row
    const size_t r = (size_t)(row0 + c * ROWS + wave);
    float* lpr = lp + r * S_EXT;
    for (int s2 = lane; s2 < S_EXT; s2 += 32)
      lpr[s2] = row[ext[s2]] - lz;

    __syncthreads();   // all reads of tile[buf] done before it is re-filled
  }
}

// ---------------------------------------------------------------------------
// Kernel B: alpha recursion. One block per batch element, 320 threads
// (tid == s for s < 257). Double-buffered alpha in LDS, one barrier per step,
// lp row for t+1 prefetched while step t computes (hides L2/HBM latency).
// ---------------------------------------------------------------------------
#define APT 320

__global__ void __launch_bounds__(APT)
ctc_alpha_kernel(const int* __restrict__ labels,
                 const int* __restrict__ lengths,
                 const int* __restrict__ lab_lens,
                 const float* __restrict__ lp,
                 float* __restrict__ costs) {
  __shared__ float alpha[2][S_EXT];
  __shared__ int ext[S_EXT];
  __shared__ unsigned char skipm[S_EXT];

  const int b   = blockIdx.x;
  const int tid = threadIdx.x;
  const float* lpb = lp + (size_t)b * Tt * S_EXT;

  for (int s = tid; s < S_EXT; s += APT)
    ext[s] = (s & 1) ? labels[b * LMAXc + (s >> 1)] : BLANK;
  __syncthreads();
  for (int s = tid; s < S_EXT; s += APT)
    skipm[s] = (s >= 2) && (ext[s] != BLANK) && (ext[s] != ext[s - 2]);

  const int len  = lengths[b];
  const int send = 2 * lab_lens[b];

  float x0 = (tid < S_EXT) ? lpb[tid] : 0.0f;        // t = 0 row
  __syncthreads();                                    // skipm visible
  if (tid < S_EXT)
    alpha[0][tid] = (tid < 2) ? x0 : NEGF;
  __syncthreads();

  int cur = 0;
  float xn = (tid < S_EXT && len > 1) ? lpb[(size_t)S_EXT + tid] : 0.0f;
  for (int t = 1; t < len; ++t) {
    const float xc = xn;
    if (tid < S_EXT && (t + 1) < len)                // prefetch next timestep
      xn = lpb[(size_t)(t + 1) * S_EXT + tid];
    if (tid < S_EXT) {
      const float a0 = alpha[cur][tid];
      const float a1 = (tid >= 1) ? alpha[cur][tid - 1] : NEGF;
      const float a2 = skipm[tid] ? alpha[cur][tid - 2] : NEGF;
      alpha[cur ^ 1][tid] = xc + lae2(lae2(a0, a1), a2);
    }
    __syncthreads();
    cur ^= 1;
  }

  if (tid == 0)
    costs[b] = -lae2(alpha[cur][send], alpha[cur][send - 1]);
}

// ---------------------------------------------------------------------------
// Kernel C: mean over batch -> d_out[0]
// ---------------------------------------------------------------------------
__global__ void ctc_reduce_kernel(const float* __restrict__ costs,
                                  float* __restrict__ out) {
  float v = (threadIdx.x < Bb) ? costs[threadIdx.x] : 0.0f;
  for (int off = 16; off; off >>= 1) v += __shfl_xor(v, off, 32);
  if (threadIdx.x == 0) out[0] = v / (float)Bb;
}

extern "C" void kernel_launch(void* const* d_in, const int* in_sizes, int n_in,
                              void* d_out, int out_size, void* d_ws, size_t ws_size,
                              hipStream_t stream) {
  (void)in_sizes; (void)n_in; (void)out_size; (void)ws_size;
  const float* act      = (const float*)d_in[0];
  const int*   labels   = (const int*)d_in[1];
  const int*   lengths  = (const int*)d_in[2];
  const int*   lab_lens = (const int*)d_in[3];

  // workspace: lp[B*T*S] floats (~33.7 MB), then costs[B]
  float* lp    = (float*)d_ws;
  float* costs = lp + (size_t)Bb * Tt * S_EXT;

  ctc_logz_lp_kernel<<<LOGZ_BLOCKS, 128, 0, stream>>>(act, labels, lp);
  ctc_alpha_kernel<<<Bb, APT, 0, stream>>>(labels, lengths, lab_lens, lp, costs);
  ctc_reduce_kernel<<<1, 32, 0, stream>>>(costs, (float*)d_out);
}